// MultiHeadHyperbolicAttention_11390253269606
// MI455X (gfx1250) — compile-verified
//
#include <hip/hip_runtime.h>
#include <hip/hip_bf16.h>

typedef __attribute__((ext_vector_type(2))) float v2f;
typedef __attribute__((ext_vector_type(8))) float v8f;

#define Bdim 2
#define Sdim 2048
#define Edim 256
#define Hdim 8
#define Ddim 32

// ---------------------------------------------------------------------------
// GEMM: out = X[M,256] @ W[256,256] + bias, via V_WMMA_F32_16X16X4_F32.
// One wave per 16x16 output tile; two interleaved accumulators break the
// WMMA->WMMA RAW chain. HEADSPLIT=1 scatters columns into [B,H,S,D].
// ---------------------------------------------------------------------------
template <int HEADSPLIT>
__global__ void proj16_kernel(const float* __restrict__ X,
                              const float* __restrict__ W,
                              const float* __restrict__ bias,
                              float* __restrict__ out) {
    const int lane = threadIdx.x & 31;
    const int wv = (blockIdx.x * blockDim.x + threadIdx.x) >> 5;
    const int tilesN = Edim / 16;                 // 16
    const int tm = wv / tilesN;
    const int tn = wv - tm * tilesN;
    const int rbase = tm * 16;
    const int nbase = tn * 16;
    const int hi = lane >> 4;
    const int lm = lane & 15;
    const int n = nbase + lm;

    v8f acc0 = {}, acc1 = {};
    const float* __restrict__ xrow = X + (size_t)(rbase + lm) * Edim;
#pragma unroll 4
    for (int k = 0; k < Edim; k += 8) {
        const int ka0 = k + 2 * hi;
        const int ka1 = k + 4 + 2 * hi;
        const float2 av0 = *(const float2*)(xrow + ka0);
        const float2 av1 = *(const float2*)(xrow + ka1);
        v2f a0; a0[0] = av0.x; a0[1] = av0.y;
        v2f a1; a1[0] = av1.x; a1[1] = av1.y;
        v2f b0; b0[0] = W[(size_t)ka0 * Edim + n];
                b0[1] = W[(size_t)(ka0 + 1) * Edim + n];
        v2f b1; b1[0] = W[(size_t)ka1 * Edim + n];
                b1[1] = W[(size_t)(ka1 + 1) * Edim + n];
        acc0 = __builtin_amdgcn_wmma_f32_16x16x4_f32(
            false, a0, false, b0, (short)0, acc0, false, false);
        acc1 = __builtin_amdgcn_wmma_f32_16x16x4_f32(
            false, a1, false, b1, (short)0, acc1, false, false);
    }
    const float bv = bias[n];
#pragma unroll
    for (int i = 0; i < 8; ++i) {
        const int m = i + 8 * hi;
        const int r = rbase + m;
        const float v = acc0[i] + acc1[i] + bv;
        if (HEADSPLIT) {
            const int bb = r >> 11;               // r / S
            const int s  = r & (Sdim - 1);
            const int h  = n >> 5;                // n / D
            const int d  = n & (Ddim - 1);
            out[(((size_t)bb * Hdim + h) * Sdim + s) * Ddim + d] = v;
        } else {
            out[(size_t)r * Edim + n] = v;
        }
    }
}

// ---------------------------------------------------------------------------
// Per-row squared norms of Q and K ([B*H*S] rows of 32).
// ---------------------------------------------------------------------------
__global__ void rownorm_kernel(const float* __restrict__ Q,
                               const float* __restrict__ K,
                               float* __restrict__ qq,
                               float* __restrict__ kk) {
    const int idx = blockIdx.x * blockDim.x + threadIdx.x;
    if (idx >= Bdim * Hdim * Sdim) return;
    const float4* q4 = (const float4*)(Q + (size_t)idx * Ddim);
    const float4* k4 = (const float4*)(K + (size_t)idx * Ddim);
    float sq = 0.f, sk = 0.f;
#pragma unroll
    for (int t = 0; t < 8; ++t) {
        float4 a = q4[t], b = k4[t];
        sq += a.x * a.x + a.y * a.y + a.z * a.z + a.w * a.w;
        sk += b.x * b.x + b.y * b.y + b.z * b.z + b.w * b.w;
    }
    qq[idx] = sq;
    kk[idx] = sk;
}

// Poincare weight with c=1:  w = exp(-2*atanh(x)) = (1-x)/(1+x)
__device__ __forceinline__ float poincare_w(float s, float qqv, float kkv) {
    const float uy = -s;
    const float A  = 1.0f + 2.0f * uy + kkv;
    const float Bc = 1.0f - qqv;
    const float num2 = A * A * qqv + 2.0f * A * Bc * uy + Bc * Bc * kkv;
    const float den  = 1.0f + 2.0f * uy + qqv * kkv;
    const float nrm  = sqrtf(fmaxf(num2, 0.0f)) / fmaxf(fabsf(den), 1e-15f);
    const float x    = fminf(nrm, 1.0f - 1e-5f);
    return (1.0f - x) / (1.0f + x);
}

// ---------------------------------------------------------------------------
// Pass 2a: per-(b,h,qtile) row sums of w (no stores of w).
// One wave per 16-row q tile; two k-tiles per iteration for WMMA ILP.
// ---------------------------------------------------------------------------
__global__ void attn_rowsum_kernel(const float* __restrict__ Q,
                                   const float* __restrict__ K,
                                   const float* __restrict__ qq,
                                   const float* __restrict__ kk,
                                   float* __restrict__ invsum) {
    const int lane = threadIdx.x & 31;
    const int wv = (blockIdx.x * blockDim.x + threadIdx.x) >> 5;
    const int qt = wv & (Sdim / 16 - 1);          // 128 q tiles
    const int bh = wv >> 7;
    const int qbase = qt * 16;
    const int hi = lane >> 4;
    const int lm = lane & 15;

    const float* __restrict__ Qb  = Q  + (size_t)bh * Sdim * Ddim;
    const float* __restrict__ Kb  = K  + (size_t)bh * Sdim * Ddim;
    const float* __restrict__ qqb = qq + (size_t)bh * Sdim;
    const float* __restrict__ kkb = kk + (size_t)bh * Sdim;

    v2f afr[8];
    const float* qrow = Qb + (size_t)(qbase + lm) * Ddim;
#pragma unroll
    for (int t = 0; t < 8; ++t) {
        const float2 t2 = *(const float2*)(qrow + 4 * t + 2 * hi);
        afr[t][0] = t2.x; afr[t][1] = t2.y;
    }
    float qqv[8], rs[8];
#pragma unroll
    for (int i = 0; i < 8; ++i) { qqv[i] = qqb[qbase + i + 8 * hi]; rs[i] = 0.f; }

    for (int kb = 0; kb < Sdim; kb += 32) {
        if (kb + 32 < Sdim)
            __builtin_prefetch(Kb + (size_t)(kb + 32 + lm) * Ddim, 0, 3);
        const float kkv0 = kkb[kb + lm];
        const float kkv1 = kkb[kb + 16 + lm];
        const float* krow0 = Kb + (size_t)(kb + lm) * Ddim;
        const float* krow1 = Kb + (size_t)(kb + 16 + lm) * Ddim;
        v8f c0 = {}, c1 = {};
#pragma unroll
        for (int t = 0; t < 8; ++t) {
            const float2 kv0 = *(const float2*)(krow0 + 4 * t + 2 * hi);
            const float2 kv1 = *(const float2*)(krow1 + 4 * t + 2 * hi);
            v2f b0; b0[0] = kv0.x; b0[1] = kv0.y;
            v2f b1; b1[0] = kv1.x; b1[1] = kv1.y;
            c0 = __builtin_amdgcn_wmma_f32_16x16x4_f32(
                false, afr[t], false, b0, (short)0, c0, false, false);
            c1 = __builtin_amdgcn_wmma_f32_16x16x4_f32(
                false, afr[t], false, b1, (short)0, c1, false, false);
        }
#pragma unroll
        for (int i = 0; i < 8; ++i) {
            rs[i] += poincare_w(c0[i], qqv[i], kkv0);
            rs[i] += poincare_w(c1[i], qqv[i], kkv1);
        }
    }
#pragma unroll
    for (int off = 1; off < 16; off <<= 1)
#pragma unroll
        for (int i = 0; i < 8; ++i)
            rs[i] += __shfl_xor(rs[i], off, 16);
    if (lm == 0) {
        float* inv = invsum + (size_t)bh * Sdim + qbase + 8 * hi;
#pragma unroll
        for (int i = 0; i < 8; ++i) inv[i] = 1.0f / rs[i];
    }
}

// ---------------------------------------------------------------------------
// Pass 2b: recompute w tiles, normalize, stream-store w, and accumulate
// out = w_norm @ V via WMMA (w transposed lane-wise through LDS).
// Two k-tiles per iteration; LDS row stride 34 holds both padded tiles.
// ---------------------------------------------------------------------------
__global__ void attn_wv_kernel(const float* __restrict__ Q,
                               const float* __restrict__ K,
                               const float* __restrict__ V,
                               const float* __restrict__ qq,
                               const float* __restrict__ kk,
                               const float* __restrict__ invsum,
                               float* __restrict__ w_out,
                               float* __restrict__ O) {
    __shared__ float lds[4 * 16 * 34];            // 4 waves/block, 16x34 each
    const int lane = threadIdx.x & 31;
    const int wv = (blockIdx.x * blockDim.x + threadIdx.x) >> 5;
    const int qt = wv & (Sdim / 16 - 1);
    const int bh = wv >> 7;
    const int qbase = qt * 16;
    const int hi = lane >> 4;
    const int lm = lane & 15;
    float* myLds = lds + (threadIdx.x >> 5) * (16 * 34);

    const float* __restrict__ Qb  = Q  + (size_t)bh * Sdim * Ddim;
    const float* __restrict__ Kb  = K  + (size_t)bh * Sdim * Ddim;
    const float* __restrict__ Vb  = V  + (size_t)bh * Sdim * Ddim;
    const float* __restrict__ qqb = qq + (size_t)bh * Sdim;
    const float* __restrict__ kkb = kk + (size_t)bh * Sdim;
    const float* __restrict__ ivb = invsum + (size_t)bh * Sdim;

    v2f afr[8];
    const float* qrow = Qb + (size_t)(qbase + lm) * Ddim;
#pragma unroll
    for (int t = 0; t < 8; ++t) {
        const float2 t2 = *(const float2*)(qrow + 4 * t + 2 * hi);
        afr[t][0] = t2.x; afr[t][1] = t2.y;
    }
    float qqv[8], invq[8];
#pragma unroll
    for (int i = 0; i < 8; ++i) {
        qqv[i]  = qqb[qbase + i + 8 * hi];
        invq[i] = ivb[qbase + i + 8 * hi];
    }

    v8f oacc0 = {}, oacc1 = {};
    float* __restrict__ wrow = w_out + ((size_t)bh * Sdim + qbase) * Sdim;

    for (int kb = 0; kb < Sdim; kb += 32) {
        if (kb + 32 < Sdim)
            __builtin_prefetch(Kb + (size_t)(kb + 32 + lm) * Ddim, 0, 3);
        const float kkv0 = kkb[kb + lm];
        const float kkv1 = kkb[kb + 16 + lm];
        const float* krow0 = Kb + (size_t)(kb + lm) * Ddim;
        const float* krow1 = Kb + (size_t)(kb + 16 + lm) * Ddim;
        v8f c0 = {}, c1 = {};
#pragma unroll
        for (int t = 0; t < 8; ++t) {
            const float2 kv0 = *(const float2*)(krow0 + 4 * t + 2 * hi);
            const float2 kv1 = *(const float2*)(krow1 + 4 * t + 2 * hi);
            v2f b0; b0[0] = kv0.x; b0[1] = kv0.y;
            v2f b1; b1[0] = kv1.x; b1[1] = kv1.y;
            c0 = __builtin_amdgcn_wmma_f32_16x16x4_f32(
                false, afr[t], false, b0, (short)0, c0, false, false);
            c1 = __builtin_amdgcn_wmma_f32_16x16x4_f32(
                false, afr[t], false, b1, (short)0, c1, false, false);
        }
#pragma unroll
        for (int i = 0; i < 8; ++i) {
            const int m = i + 8 * hi;
            const float wn0 = poincare_w(c0[i], qqv[i], kkv0) * invq[i];
            const float wn1 = poincare_w(c1[i], qqv[i], kkv1) * invq[i];
            __builtin_nontemporal_store(wn0, wrow + (size_t)m * Sdim + kb + lm);
            __builtin_nontemporal_store(wn1, wrow + (size_t)m * Sdim + kb + 16 + lm);
            myLds[m * 34 + lm]      = wn0;
            myLds[m * 34 + 17 + lm] = wn1;
        }
        __syncthreads();
        // out_tile[16x32] += w[16x32] @ V[32x32]; per half: 16-k contraction
#pragma unroll
        for (int half = 0; half < 2; ++half) {
            const float* ldsb = myLds + 17 * half;
            const int kbb = kb + 16 * half;
#pragma unroll
            for (int t = 0; t < 4; ++t) {
                v2f a; a[0] = ldsb[lm * 34 + 4 * t + 2 * hi];
                       a[1] = ldsb[lm * 34 + 4 * t + 2 * hi + 1];
                const float* vrow0 = Vb + (size_t)(kbb + 4 * t + 2 * hi) * Ddim;
                const float* vrow1 = vrow0 + Ddim;
                v2f b0; b0[0] = vrow0[lm];      b0[1] = vrow1[lm];
                oacc0 = __builtin_amdgcn_wmma_f32_16x16x4_f32(
                    false, a, false, b0, (short)0, oacc0, false, false);
                v2f b1; b1[0] = vrow0[16 + lm]; b1[1] = vrow1[16 + lm];
                oacc1 = __builtin_amdgcn_wmma_f32_16x16x4_f32(
                    false, a, false, b1, (short)0, oacc1, false, false);
            }
        }
        __syncthreads();
    }
    // write merged pre-projection output O[b, s, h*32 + d]
    const int b = bh >> 3, h = bh & 7;
#pragma unroll
    for (int i = 0; i < 8; ++i) {
        const int s = qbase + i + 8 * hi;
        float* orow = O + ((size_t)b * Sdim + s) * Edim + h * Ddim;
        orow[lm]      = oacc0[i];
        orow[16 + lm] = oacc1[i];
    }
}

// ---------------------------------------------------------------------------
extern "C" void kernel_launch(void* const* d_in, const int* in_sizes, int n_in,
                              void* d_out, int out_size, void* d_ws, size_t ws_size,
                              hipStream_t stream) {
    const float* q_in = (const float*)d_in[0];
    const float* k_in = (const float*)d_in[1];
    const float* v_in = (const float*)d_in[2];
    const float* Wq = (const float*)d_in[3];
    const float* bq = (const float*)d_in[4];
    const float* Wk = (const float*)d_in[5];
    const float* bk = (const float*)d_in[6];
    const float* Wv = (const float*)d_in[7];
    const float* bv = (const float*)d_in[8];
    const float* Wo = (const float*)d_in[9];
    const float* bo = (const float*)d_in[10];

    float* out = (float*)d_out;                         // [B,S,E]
    float* w_out = out + (size_t)Bdim * Sdim * Edim;    // [B,H,S,S]

    const size_t nQ = (size_t)Bdim * Hdim * Sdim * Ddim;    // 1M floats
    float* ws = (float*)d_ws;
    float* Q   = ws;
    float* K   = ws + nQ;
    float* V   = ws + 2 * nQ;
    float* O   = ws + 3 * nQ;                               // [B,S,E]
    float* qq  = ws + 4 * nQ;
    float* kk  = qq + (size_t)Bdim * Hdim * Sdim;
    float* inv = kk + (size_t)Bdim * Hdim * Sdim;

    const int M = Bdim * Sdim;                              // 4096
    const int projWaves = (M / 16) * (Edim / 16);           // 4096
    const dim3 projGrid(projWaves / 4), blk128(128);

    proj16_kernel<1><<<projGrid, blk128, 0, stream>>>(q_in, Wq, bq, Q);
    proj16_kernel<1><<<projGrid, blk128, 0, stream>>>(k_in, Wk, bk, K);
    proj16_kernel<1><<<projGrid, blk128, 0, stream>>>(v_in, Wv, bv, V);

    rownorm_kernel<<<dim3((Bdim * Hdim * Sdim) / 256), dim3(256), 0, stream>>>(Q, K, qq, kk);

    const int attnWaves = Bdim * Hdim * (Sdim / 16);        // 2048
    attn_rowsum_kernel<<<dim3(attnWaves / 4), blk128, 0, stream>>>(Q, K, qq, kk, inv);
    attn_wv_kernel<<<dim3(attnWaves / 4), blk128, 0, stream>>>(Q, K, V, qq, kk, inv,
                                                              w_out, O);

    proj16_kernel<0><<<projGrid, blk128, 0, stream>>>(O, Wo, bo, out);
}